// EdgeLayer_35373350650342
// MI455X (gfx1250) — compile-verified
//
#include <hip/hip_runtime.h>
#include <stdint.h>

// Min-plus "GEMM" for the EdgeLayer reference:
//   out[b,o] = min_j ( x[b,j] + bernoulli(sigmoid(e1-e0))[o,j] )
// B=32 (== wave32 lanes), IN=OUT=2048.
// Memory-bound on the 32MB edges stream (~1.4us @ 23.3TB/s).
// edges tiles arrive via TDM (tensor_load_to_lds, double-buffered);
// x tiles arrive via async global->LDS DMA (per-lane LDS addrs, x[b][jj] rows padded to 258).

#define B_SZ   32
#define IN_F   2048
#define OUT_F  2048
#define TILE_J 256
#define NTILES (IN_F / TILE_J)      // 8
#define LDSTR  258                  // ==2 mod 4 -> conflict-free ds_load_b64 on x tile
#define WAVES  4                    // waves (= output rows) per block
#define NTHREADS (WAVES * 32)

typedef float        f32x2 __attribute__((ext_vector_type(2)));
typedef unsigned int u32x4 __attribute__((ext_vector_type(4)));
typedef int          i32x4 __attribute__((ext_vector_type(4)));
typedef int          i32x8 __attribute__((ext_vector_type(8)));
typedef int          v2i   __attribute__((__vector_size__(2 * sizeof(int))));

typedef __attribute__((address_space(1))) int* gint_p;   // global (AS1) int
typedef __attribute__((address_space(3))) int* lint_p;   // LDS (AS3) int
typedef __attribute__((address_space(1))) v2i* g2i_p;    // global (AS1) v2i32
typedef __attribute__((address_space(3))) v2i* l2i_p;    // LDS (AS3) v2i32

#define HAVE_TDM __has_builtin(__builtin_amdgcn_tensor_load_to_lds)

__device__ __forceinline__ uint32_t hash_u32(uint32_t v) {
  v ^= v >> 16; v *= 0x7feb352dU;
  v ^= v >> 15; v *= 0x846ca68bU;
  v ^= v >> 16;
  return v;
}

__device__ __forceinline__ uint32_t ballot32(bool p) {
#if __has_builtin(__builtin_amdgcn_ballot_w32)
  return __builtin_amdgcn_ballot_w32(p);
#else
  return (uint32_t)__ballot(p);
#endif
}

// async global -> LDS copy of 2 dwords (per-lane LDS dest address)
__device__ __forceinline__ void async_copy_qw(const float* g, float* l) {
#if __has_builtin(__builtin_amdgcn_global_load_async_to_lds_b64)
  __builtin_amdgcn_global_load_async_to_lds_b64((g2i_p)g, (l2i_p)l, 0, 0);
#elif __has_builtin(__builtin_amdgcn_global_load_async_to_lds_b32)
  __builtin_amdgcn_global_load_async_to_lds_b32((gint_p)g,       (lint_p)l,       0, 0);
  __builtin_amdgcn_global_load_async_to_lds_b32((gint_p)(g + 1), (lint_p)(l + 1), 0, 0);
#else
  l[0] = g[0]; l[1] = g[1];
#endif
}

__device__ __forceinline__ void async_wait_all() {
#if __has_builtin(__builtin_amdgcn_s_wait_asynccnt)
  __builtin_amdgcn_s_wait_asynccnt(0);
#else
  asm volatile("s_wait_asynccnt 0" ::: "memory");
#endif
}

template <int N>
__device__ __forceinline__ void tensor_wait() {
#if __has_builtin(__builtin_amdgcn_s_wait_tensorcnt)
  __builtin_amdgcn_s_wait_tensorcnt(N);
#else
  asm volatile("s_wait_tensorcnt %0" :: "i"(N) : "memory");
#endif
}

#if HAVE_TDM
// Issue a TDM load of one contiguous 2KB edge tile (256 x 8B) into LDS.
// D# group0: count=1 | lds_addr | global_addr | type=2
// D# group1: data_size=3(8B), tensor_dim0=4M elems, tile_dim0=256, 1-D tile.
__device__ __forceinline__ void tdm_load_edge_tile(uint64_t gaddr, uint32_t lds_addr) {
  u32x4 g0;
  g0.x = 1u;                                                  // count=1, user D#
  g0.y = lds_addr;                                            // lds_addr[31:0]
  g0.z = (uint32_t)gaddr;                                     // global_addr[31:0]
  g0.w = ((uint32_t)(gaddr >> 32) & 0x01FFFFFFu) | (2u << 30);// ga[56:32] | type=2

  const uint32_t td0 = (uint32_t)IN_F * (uint32_t)OUT_F;      // 4,194,304 8B elems
  i32x8 g1;
  g1[0] = (int)(3u << 16);                                    // data_size=3 (8 bytes)
  g1[1] = (int)((td0 & 0xFFFFu) << 16);                       // tensor_dim0[15:0]
  g1[2] = (int)((td0 >> 16) | (1u << 16));                    // tensor_dim0[31:16] | tensor_dim1=1
  g1[3] = (int)((uint32_t)TILE_J << 16);                      // tile_dim0=256 (tile_dim1=0 -> 1D)
  g1[4] = 0; g1[5] = 0; g1[6] = 0; g1[7] = 0;                 // dims/strides unused

  i32x4 z4 = {0, 0, 0, 0};
  i32x8 z8 = {0, 0, 0, 0, 0, 0, 0, 0};
  __builtin_amdgcn_tensor_load_to_lds(g0, g1, z4, z4, z8, 0); // 6-arg toolchain variant
}
#endif

__global__ __launch_bounds__(NTHREADS, 1) void edge_minplus_kernel(
    const float* __restrict__ x,
    const float* __restrict__ edges,
    float* __restrict__ out)
{
  __shared__ float xT[B_SZ * LDSTR];          // 33,024 B: x[b][jj], stride 258 dwords
#if HAVE_TDM
  __shared__ f32x2 eT[2][WAVES][TILE_J];      // 16,384 B: double-buffered edge tiles
#endif

  const int tid  = (int)threadIdx.x;
  const int lane = tid & 31;                  // role 1: j-offset (mask gen); role 2: batch b
  const int wv   = tid >> 5;
  const int o    = (int)blockIdx.x * WAVES + wv;   // one wave per output row

  const uint32_t rc = hash_u32(0x9E3779B9u ^ (uint32_t)o) & (IN_F - 1);

  float acc = 3.402823466e+38f;
  uint32_t cnt = 0;

  const f32x2* __restrict__ erow = (const f32x2*)(edges + (size_t)o * IN_F * 2);

#if HAVE_TDM
  const uint32_t eT_base = (uint32_t)(uintptr_t)(lint_p)&eT[0][0][0];
  const uint64_t eg_base = (uint64_t)(uintptr_t)erow;
  // prologue: kick off DMA for tile 0
  tdm_load_edge_tile(eg_base, eT_base + (uint32_t)(wv * TILE_J * 8));
#endif

  for (int t = 0; t < NTILES; ++t) {
    const int j0 = t * TILE_J;

    // ---- stage x[:, j0:j0+256] via async global->LDS (overlaps the TDM) ----
    #pragma unroll 4
    for (int b = 0; b < B_SZ; ++b) {
      async_copy_qw(x + b * IN_F + j0 + 2 * tid, &xT[b * LDSTR + 2 * tid]);
    }

#if HAVE_TDM
    // kick off DMA for tile t+1 into the other buffer
    if (t + 1 < NTILES) {
      tdm_load_edge_tile(eg_base + (uint64_t)(t + 1) * (TILE_J * 8),
                         eT_base + (uint32_t)(((((t + 1) & 1) * WAVES) + wv) * TILE_J * 8));
    }
#endif

    async_wait_all();
    __syncthreads();                          // x tile visible to all waves

#if HAVE_TDM
    if (t + 1 < NTILES) tensor_wait<1>();     // tile t complete (in-order TENSORcnt)
    else                tensor_wait<0>();
    const f32x2* etile = &eT[t & 1][wv][0];
#endif

    if (j0 + TILE_J < IN_F)
      __builtin_prefetch((const void*)(erow + (j0 + TILE_J)), 0, 0);

    // ---- 8 chunks of 32 j: lanes-as-j make mask bits, ballot shares them ----
    for (int c = 0; c < TILE_J / 32; ++c) {
      const int j = j0 + c * 32 + lane;
#if HAVE_TDM
      const f32x2 e = etile[c * 32 + lane];   // ds_load_b64, conflict-free
#else
      const f32x2 e = __builtin_nontemporal_load(erow + j);
#endif
      const float d = e.y - e.x;              // softmax-over-2 == sigmoid(e1-e0)
      const float p = 1.0f / (1.0f + __expf(-d));
      const uint32_t h = hash_u32((uint32_t)(o * IN_F + j));
      const float u = (float)(h >> 8) * (1.0f / 16777216.0f);
      const uint32_t bal = ballot32(u < p);   // 32 bernoulli bits in an SGPR
      cnt += (uint32_t)__builtin_popcount(bal);

      const f32x2* xrow = (const f32x2*)&xT[lane * LDSTR + c * 32];
      #pragma unroll
      for (int q = 0; q < 16; ++q) {
        const f32x2 xv = xrow[q];             // conflict-free ds_load_b64
        const float m0 = ((bal >> (2 * q))     & 1u) ? 1.0f : 0.0f;
        const float m1 = ((bal >> (2 * q + 1)) & 1u) ? 1.0f : 0.0f;
        acc = fminf(acc, xv.x + m0);
        acc = fminf(acc, xv.y + m1);
      }
    }
    __syncthreads();                          // protect xT (and eT bufs) for next round
  }

  // ---- zero-connection fix (wave-uniform, astronomically rare path) ----
  if (cnt == 0u) {
    acc = 3.402823466e+38f;
    for (int j = 0; j < IN_F; ++j) {
      const float xv = x[lane * IN_F + j];    // L2-resident by now
      acc = fminf(acc, xv + (((uint32_t)j == rc) ? 1.0f : 0.0f));
    }
  }

  out[lane * OUT_F + o] = acc;                // lane == batch b
}

extern "C" void kernel_launch(void* const* d_in, const int* in_sizes, int n_in,
                              void* d_out, int out_size, void* d_ws, size_t ws_size,
                              hipStream_t stream) {
  const float* x     = (const float*)d_in[0];   // [32, 2048] fp32
  const float* edges = (const float*)d_in[1];   // [2048, 2048, 2] fp32
  float* out         = (float*)d_out;           // [32, 2048] fp32

  dim3 grid(OUT_F / WAVES);   // 512 blocks, 4 waves (=4 output rows) each
  dim3 block(NTHREADS);
  hipLaunchKernelGGL(edge_minplus_kernel, grid, block, 0, stream, x, edges, out);

  (void)in_sizes; (void)n_in; (void)out_size; (void)d_ws; (void)ws_size;
}